// GestaltNet_7387343749542
// MI455X (gfx1250) — compile-verified
//
#include <hip/hip_runtime.h>
#include <hip/hip_bf16.h>
#include <stdint.h>

// ---------------------------------------------------------------------------
// GestaltNet RPN+keypoint pipeline for MI455X (gfx1250, wave32).
// Memory-bound op (~90MB HBM traffic, heatmap is L2-resident) — no GEMM
// content, so WMMA is inapplicable. CDNA5-specific path exercised: async
// global->LDS (global_load_async_to_lds_b128, ASYNCcnt) with a double-buffered
// pipeline: issue row x+1, s_wait_asynccnt 1, consume row x.
// ---------------------------------------------------------------------------

#define IMGSZ   1024
#define NANCH   (256 * 256 * 9)   // 589824
#define NPRE    600
#define NPOST   6
#define NJ      16
#define NMS_T   0.7f
#define MINSZ   16.0f

#define USE_ASYNC_LDS 1

struct Ctrl {
  unsigned prefix;
  unsigned K;
  unsigned T;
  unsigned count_gt;
  unsigned eq_needed;
  unsigned cntA;
  unsigned cntB;
  unsigned pad_;
  unsigned hist[256];
};

// Monotone float -> uint mapping (larger float => larger uint).
__device__ __forceinline__ unsigned ordkey(float f) {
  unsigned b = __float_as_uint(f);
  return (b & 0x80000000u) ? ~b : (b | 0x80000000u);
}

// Reference bbox_calculate + clip.
__device__ __forceinline__ float4 decode_clip(const float* __restrict__ anchor,
                                              const float* __restrict__ shift,
                                              int i) {
  float a0 = anchor[4 * i + 0], a1 = anchor[4 * i + 1];
  float a2 = anchor[4 * i + 2], a3 = anchor[4 * i + 3];
  float t0 = shift[4 * i + 0], t1 = shift[4 * i + 1];
  float t2 = shift[4 * i + 2], t3 = shift[4 * i + 3];
  float aw = a2 - a0, ah = a3 - a1;
  float ax = a0 + 0.5f * aw, ay = a1 + 0.5f * ah;
  float bx = ax + aw * t0, by = ay + ah * t1;
  float bw = aw * expf(t2), bh = ah * expf(t3);
  float xmin = bx - 0.5f * bw, ymin = by - 0.5f * bh;
  float xmax = bw + xmin, ymax = bh + ymin;
  xmin = fminf(fmaxf(xmin, 0.f), (float)IMGSZ);
  ymin = fminf(fmaxf(ymin, 0.f), (float)IMGSZ);
  xmax = fminf(fmaxf(xmax, 0.f), (float)IMGSZ);
  ymax = fminf(fmaxf(ymax, 0.f), (float)IMGSZ);
  return make_float4(xmin, ymin, xmax, ymax);
}

// Stage 1: sortable key per anchor. Softmax fg is monotone in (s1-s0) and the
// score value itself is never used downstream (only ordering), so we sort on
// ordkey(s1-s0); invalid boxes get key 0 (below every finite value's key).
__global__ void k_scores(const float* __restrict__ shift,
                         const float* __restrict__ score,
                         const float* __restrict__ anchor,
                         unsigned* __restrict__ ukeys) {
  int i = blockIdx.x * blockDim.x + threadIdx.x;
  if (i >= NANCH) return;
  float4 b = decode_clip(anchor, shift, i);
  bool valid = ((b.z - b.x) >= MINSZ) && ((b.w - b.y) >= MINSZ);
  unsigned u = 0u;
  if (valid) u = ordkey(score[2 * i + 1] - score[2 * i + 0]);
  ukeys[i] = u;
}

__global__ void k_init(Ctrl* c) {
  int t = threadIdx.x;
  if (t == 0) {
    c->prefix = 0u; c->K = NPRE; c->T = 0u;
    c->count_gt = 0u; c->eq_needed = 0u; c->cntA = 0u; c->cntB = 0u;
  }
  c->hist[t] = 0u;
}

// Radix-select histogram pass (byte `sh`, prefix-matched on bits in himask).
__global__ void k_hist(const unsigned* __restrict__ ukeys, Ctrl* c,
                       int sh, unsigned himask) {
  __shared__ unsigned h[256];
  int t = threadIdx.x;
  h[t] = 0u;
  __syncthreads();
  unsigned pfx = c->prefix & himask;
  for (int i = blockIdx.x * blockDim.x + t; i < NANCH;
       i += gridDim.x * blockDim.x) {
    unsigned u = ukeys[i];
    if ((u & himask) == pfx) atomicAdd(&h[(u >> sh) & 255u], 1u);
  }
  __syncthreads();
  if (h[t]) atomicAdd(&c->hist[t], h[t]);
}

// Pick the bin containing the K-th largest element; narrow prefix.
__global__ void k_resolve(Ctrl* c, int sh, int last) {
  if (threadIdx.x != 0 || blockIdx.x != 0) return;
  unsigned K = c->K;
  unsigned acc = 0, above = 0;
  int chosen = 0;
  for (int b = 255; b >= 0; --b) {
    unsigned hb = c->hist[b];
    if (acc + hb >= K) { chosen = b; above = acc; break; }
    acc += hb;
  }
  c->prefix |= ((unsigned)chosen) << sh;
  c->K = K - above;
  for (int b = 0; b < 256; ++b) c->hist[b] = 0u;
  if (last) {
    c->T = c->prefix;
    c->eq_needed = c->K;             // # elements == T to take (lowest indices)
    c->count_gt = NPRE - c->K;       // # elements strictly > T
    c->cntA = 0u; c->cntB = 0u;
  }
}

// Gather (key,index) for u > T and indices for u == T.
__global__ void k_compact(const unsigned* __restrict__ ukeys, Ctrl* c,
                          unsigned long long* __restrict__ listA,
                          unsigned* __restrict__ listB) {
  unsigned T = c->T;
  for (int i = blockIdx.x * blockDim.x + threadIdx.x; i < NANCH;
       i += gridDim.x * blockDim.x) {
    unsigned u = ukeys[i];
    if (u > T) {
      unsigned p = atomicAdd(&c->cntA, 1u);
      if (p < 1024u)
        listA[p] = ((unsigned long long)u << 32) | (unsigned)~(unsigned)i;
    } else if (u == T) {
      unsigned p = atomicAdd(&c->cntB, 1u);
      if (p < 4096u) listB[p] = (unsigned)i;
    }
  }
}

// Sort candidates: A descending by (score, then ascending index via ~idx in
// the low word), B ascending by index; emit order[600]; decode top-600 boxes.
__global__ void __launch_bounds__(1024)
k_sortmerge(Ctrl* c, const unsigned long long* __restrict__ listA,
            const unsigned* __restrict__ listB, unsigned* __restrict__ order,
            const float* __restrict__ anchor, const float* __restrict__ shift,
            float4* __restrict__ boxes600) {
  __shared__ unsigned long long ka[1024];
  __shared__ unsigned kb[4096];
  int t = threadIdx.x;
  unsigned nA = c->cntA; if (nA > 1024u) nA = 1024u;
  unsigned nB = c->cntB; if (nB > 4096u) nB = 4096u;
  ka[t] = (t < (int)nA) ? listA[t] : 0ull;
  for (int i = t; i < 4096; i += 1024)
    kb[i] = (i < (int)nB) ? listB[i] : 0xFFFFFFFFu;
  __syncthreads();
  // Bitonic ascending sort of ka[1024] (pad 0 sinks to the bottom).
  for (int k = 2; k <= 1024; k <<= 1) {
    for (int jj = k >> 1; jj > 0; jj >>= 1) {
      int ixj = t ^ jj;
      if (ixj > t) {
        unsigned long long a = ka[t], b2 = ka[ixj];
        bool up = ((t & k) == 0);
        if ((a > b2) == up) { ka[t] = b2; ka[ixj] = a; }
      }
      __syncthreads();
    }
  }
  // Bitonic ascending sort of kb[4096], 4 virtual lanes per thread.
  for (int k = 2; k <= 4096; k <<= 1) {
    for (int jj = k >> 1; jj > 0; jj >>= 1) {
      for (int v = t; v < 4096; v += 1024) {
        int ixj = v ^ jj;
        if (ixj > v) {
          unsigned a = kb[v], b2 = kb[ixj];
          bool up = ((v & k) == 0);
          if ((a > b2) == up) { kb[v] = b2; kb[ixj] = a; }
        }
      }
      __syncthreads();
    }
  }
  unsigned cg = c->count_gt;
  unsigned en = c->eq_needed;
  if (t < (int)cg && t < NPRE)
    order[t] = ~(unsigned)ka[1023 - t];       // descending by score
  else if (t < (int)(cg + en) && t < NPRE)
    order[t] = kb[t - cg];                    // ties: ascending index
  __syncthreads();
  if (t < NPRE) boxes600[t] = decode_clip(anchor, shift, (int)order[t]);
}

// Sequential greedy NMS (exact reference semantics) + init argmax slots.
__global__ void __launch_bounds__(1024)
k_nms(const float4* __restrict__ boxes600, int* __restrict__ roi6,
      unsigned long long* __restrict__ slots) {
  __shared__ float4 sb[NPRE];
  __shared__ unsigned char supp[NPRE];
  int t = threadIdx.x;
  if (t < NPRE) { sb[t] = boxes600[t]; supp[t] = 0; }
  if (t < NPOST * NJ) slots[t] = 0x00000000FFFFFFFFull;  // u=0, flat=~0 -> 0
  __syncthreads();
  for (int i = 0; i < NPRE; ++i) {
    bool si = (supp[i] != 0);
    float4 ci = sb[i];
    if (!si && t > i && t < NPRE) {
      float4 bb = sb[t];
      float ix1 = fmaxf(ci.x, bb.x), iy1 = fmaxf(ci.y, bb.y);
      float ix2 = fminf(ci.z, bb.z), iy2 = fminf(ci.w, bb.w);
      float inter = fmaxf(ix2 - ix1, 0.f) * fmaxf(iy2 - iy1, 0.f);
      float ai = (ci.z - ci.x) * (ci.w - ci.y);
      float at = (bb.z - bb.x) * (bb.w - bb.y);
      float iou = inter / (ai + at - inter);
      if (iou > NMS_T) supp[t] = 1;
    }
    __syncthreads();
  }
  if (t == 0) {
    int kidx[NPOST] = {0, 0, 0, 0, 0, 0};   // nonzero(..., fill_value=0)
    int cnt = 0;
    for (int i = 0; i < NPRE && cnt < NPOST; ++i)
      if (!supp[i]) kidx[cnt++] = i;
    for (int r = 0; r < NPOST; ++r) {
      float4 bb = sb[kidx[r]];
      roi6[4 * r + 0] = (int)bb.x; roi6[4 * r + 1] = (int)bb.y;
      roi6[4 * r + 2] = (int)bb.z; roi6[4 * r + 3] = (int)bb.w;
    }
  }
}

// Masked argmax per (roi, joint). Rows are staged into LDS via the CDNA5
// async global->LDS engine with a 2-deep pipeline: issue row x+1, wait
// asynccnt<=1 (async loads complete in order => row x landed), consume row x.
// First-occurrence tie-break matches jnp.argmax via ~flat in the low word.
__global__ void k_argmax(const float* __restrict__ hm,
                         const int* __restrict__ roi6,
                         unsigned long long* __restrict__ slots) {
  constexpr int CH = 16, RPC = IMGSZ / CH;  // 16 chunks x 64 rows
  int b = blockIdx.x;
  int r = b / (NJ * CH);
  int rem = b % (NJ * CH);
  int j = rem / CH;
  int ch = rem % CH;
  int x0 = roi6[4 * r + 0], y0 = roi6[4 * r + 1];
  int x1 = roi6[4 * r + 2], y1 = roi6[4 * r + 3];
  int xs = max(x0, ch * RPC), xe = min(x1, (ch + 1) * RPC);
  if (xs >= xe || y0 >= y1) return;   // uniform per block
  int t = threadIdx.x;
  __shared__ __align__(16) float rowbuf[2][1024];
  __shared__ unsigned long long red[256];
  const float* base = hm + (size_t)j * IMGSZ * IMGSZ;
  int ya = y0 & ~3;                   // 16B-aligned column start
  int col = ya + 4 * t;
  bool active = (col < y1);
  unsigned long long best = 0x00000000FFFFFFFFull;
#if USE_ASYNC_LDS
  unsigned lds0 = (unsigned)(uintptr_t)(void*)&rowbuf[0][4 * t];
  unsigned lds1 = (unsigned)(uintptr_t)(void*)&rowbuf[1][4 * t];
  // Prologue: stage first row into buffer 0.
  if (active) {
    unsigned long long ga =
        (unsigned long long)(uintptr_t)(base + (size_t)xs * IMGSZ + col);
    asm volatile("global_load_async_to_lds_b128 %0, %1, off"
                 :: "v"(lds0), "v"(ga) : "memory");
  }
#endif
  for (int x = xs; x < xe; ++x) {
    int cur = (x - xs) & 1;
    bool more = (x + 1 < xe);
#if USE_ASYNC_LDS
    if (more && active) {
      unsigned long long ga =
          (unsigned long long)(uintptr_t)(base + (size_t)(x + 1) * IMGSZ + col);
      unsigned nlds = cur ? lds0 : lds1;    // next row -> other buffer
      asm volatile("global_load_async_to_lds_b128 %0, %1, off"
                   :: "v"(nlds), "v"(ga) : "memory");
    }
    if (more) asm volatile("s_wait_asynccnt 0x1" ::: "memory");
    else      asm volatile("s_wait_asynccnt 0x0" ::: "memory");
    if (active) {
      float4 v = *(const float4*)&rowbuf[cur][4 * t];
#else
    if (active) {
      float4 v = *(const float4*)(base + (size_t)x * IMGSZ + col);
      (void)more;
#endif
      float vv[4] = {v.x, v.y, v.z, v.w};
#pragma unroll
      for (int q = 0; q < 4; ++q) {
        int cc = col + q;
        if (cc >= y0 && cc < y1) {
          unsigned long long key =
              ((unsigned long long)ordkey(vv[q]) << 32) |
              (unsigned)~(unsigned)(x * IMGSZ + cc);
          best = (key > best) ? key : best;
        }
      }
    }
  }
  red[t] = best;
  __syncthreads();
  for (int s = 128; s > 0; s >>= 1) {
    if (t < s) red[t] = (red[t + s] > red[t]) ? red[t + s] : red[t];
    __syncthreads();
  }
  if (t == 0) atomicMax(&slots[r * NJ + j], red[0]);
}

// Reference's (quirky) local-index math, including the width/height mix-up.
__global__ void k_final(const unsigned long long* __restrict__ slots,
                        const int* __restrict__ roi6,
                        float* __restrict__ out) {
  int t = threadIdx.x;
  if (t >= NPOST * NJ) return;
  int r = t / NJ;
  int x0 = roi6[4 * r + 0], y0 = roi6[4 * r + 1];
  int x1 = roi6[4 * r + 2], y1 = roi6[4 * r + 3];
  unsigned flat = ~(unsigned)(slots[t] & 0xFFFFFFFFull);
  int gx = (int)(flat / IMGSZ), gy = (int)(flat % IMGSZ);
  int m = y1 - y0;
  long long local = (long long)(gx - x0) * (long long)m + (long long)(gy - y0);
  float jx = floorf((float)local / (float)(x1 - x0));
  float jy = 0.f;
  if (m != 0) {
    long long md = local % (long long)m;     // python floor-mod
    if (md < 0) md += m;
    jy = (float)md;
  }
  out[2 * t + 0] = jx;
  out[2 * t + 1] = jy;
}

extern "C" void kernel_launch(void* const* d_in, const int* in_sizes, int n_in,
                              void* d_out, int out_size, void* d_ws,
                              size_t ws_size, hipStream_t stream) {
  const float* shift   = (const float*)d_in[0];
  const float* score   = (const float*)d_in[1];
  const float* heatmap = (const float*)d_in[2];
  const float* anchor  = (const float*)d_in[3];
  float* out = (float*)d_out;

  char* ws = (char*)d_ws;
  Ctrl* ctrl = (Ctrl*)ws;
  size_t off = 2048;
  unsigned* ukeys = (unsigned*)(ws + off);            off += (size_t)NANCH * 4;
  unsigned long long* listA = (unsigned long long*)(ws + off); off += 1024 * 8;
  unsigned* listB = (unsigned*)(ws + off);            off += 4096 * 4;
  unsigned* order = (unsigned*)(ws + off);            off += 1024 * 4;
  float4* boxes600 = (float4*)(ws + off);             off += 640 * 16;
  int* roi6 = (int*)(ws + off);                       off += 128;
  unsigned long long* slots = (unsigned long long*)(ws + off); off += 96 * 8;
  (void)in_sizes; (void)n_in; (void)out_size; (void)ws_size;

  k_scores<<<(NANCH + 255) / 256, 256, 0, stream>>>(shift, score, anchor, ukeys);
  k_init<<<1, 256, 0, stream>>>(ctrl);

  const int      shifts[4] = {24, 16, 8, 0};
  const unsigned masks[4]  = {0u, 0xFF000000u, 0xFFFF0000u, 0xFFFFFF00u};
  for (int p = 0; p < 4; ++p) {
    k_hist<<<1024, 256, 0, stream>>>(ukeys, ctrl, shifts[p], masks[p]);
    k_resolve<<<1, 1, 0, stream>>>(ctrl, shifts[p], (p == 3) ? 1 : 0);
  }
  k_compact<<<1024, 256, 0, stream>>>(ukeys, ctrl, listA, listB);
  k_sortmerge<<<1, 1024, 0, stream>>>(ctrl, listA, listB, order, anchor, shift,
                                      boxes600);
  k_nms<<<1, 1024, 0, stream>>>(boxes600, roi6, slots);
  k_argmax<<<NPOST * NJ * 16, 256, 0, stream>>>(heatmap, roi6, slots);
  k_final<<<1, 128, 0, stream>>>(slots, roi6, out);
}